// PacUnetGenerator_47519518163474
// MI455X (gfx1250) — compile-verified
//
#include <hip/hip_runtime.h>
#include <math.h>

typedef float v2f __attribute__((ext_vector_type(2)));
typedef float v8f __attribute__((ext_vector_type(8)));

#define TPB 256
#define WAVES 8

// ---------------------------------------------------------------------------
// Guide branch: Haar wavelet features (depthwise 2x2, pad (0,1),(0,1))
// x: (2,3,256,256) -> out: (2,12,256,256)
// ---------------------------------------------------------------------------
__global__ void wavelet_kernel(const float* __restrict__ x, float* __restrict__ out) {
  int idx = blockIdx.x * blockDim.x + threadIdx.x;
  const int N = 2 * 12 * 256 * 256;
  if (idx >= N) return;
  int px = idx & 255;
  int py = (idx >> 8) & 255;
  int oc = (idx >> 16) % 12;
  int b  = idx / (12 * 65536);
  int ci = oc >> 2;
  int f  = oc & 3;
  const float* src = x + ((size_t)b * 3 + ci) * 65536;
  float s00 = src[py * 256 + px];
  float s01 = (px + 1 < 256) ? src[py * 256 + px + 1] : 0.f;
  float s10 = (py + 1 < 256) ? src[(py + 1) * 256 + px] : 0.f;
  float s11 = (px + 1 < 256 && py + 1 < 256) ? src[(py + 1) * 256 + px + 1] : 0.f;
  float r;
  switch (f) {
    case 0:  r = s00 + s01 + s10 + s11; break;
    case 1:  r = s00 + s01 - s10 - s11; break;
    case 2:  r = s00 - s01 + s10 - s11; break;
    default: r = s00 - s01 - s10 + s11; break;
  }
  out[idx] = 0.5f * r;
}

// ---------------------------------------------------------------------------
// Generic small 5x5 conv, pad 2, 256x256, B=2.  Cin/Cout <= 12.
// ---------------------------------------------------------------------------
__global__ void conv5_kernel(const float* __restrict__ in, const float* __restrict__ w,
                             const float* __restrict__ bias, float* __restrict__ out,
                             int Cin, int Cout, int relu) {
  int idx = blockIdx.x * blockDim.x + threadIdx.x;
  int N = 2 * Cout * 65536;
  if (idx >= N) return;
  int px = idx & 255;
  int py = (idx >> 8) & 255;
  int oc = (idx >> 16) % Cout;
  int b  = idx / (Cout * 65536);
  float acc = bias[oc];
  for (int c = 0; c < Cin; ++c) {
    const float* src = in + ((size_t)b * Cin + c) * 65536;
    const float* wp  = w + ((size_t)oc * Cin + c) * 25;
    for (int ky = 0; ky < 5; ++ky) {
      int yy = py + ky - 2;
      if (yy < 0 || yy >= 256) continue;
      for (int kx = 0; kx < 5; ++kx) {
        int xx = px + kx - 2;
        if (xx < 0 || xx >= 256) continue;
        acc += src[yy * 256 + xx] * wp[ky * 5 + kx];
      }
    }
  }
  if (relu) acc = fmaxf(acc, 0.f);
  out[idx] = acc;
}

// ---------------------------------------------------------------------------
// PAC adapting kernel: K[b,t,y,x] = exp(-0.5 * sum_c (g_patch - g_center)^2)
// guide is the full-res (2,3,256,256) tensor; "sub" implements guide[::sub,::sub]
// ---------------------------------------------------------------------------
__global__ void pac_k_kernel(const float* __restrict__ guide, int sub, int Hg,
                             int stride, int OH, int OW, float* __restrict__ K) {
  int idx = blockIdx.x * blockDim.x + threadIdx.x;
  int N = 2 * 9 * OH * OW;
  if (idx >= N) return;
  int ox = idx % OW;
  int oy = (idx / OW) % OH;
  int t  = (idx / (OW * OH)) % 9;
  int b  = idx / (9 * OH * OW);
  int i = t / 3, j = t - 3 * i;
  int cy = oy * stride, cx = ox * stride;          // center tap (i=1,j=1)
  int py = cy + i - 1,  px = cx + j - 1;
  bool pv = (py >= 0 && py < Hg && px >= 0 && px < Hg);
  float d2 = 0.f;
  for (int c = 0; c < 3; ++c) {
    const float* g = guide + ((size_t)b * 3 + c) * 65536;
    float cv = g[(size_t)sub * (cy * 256 + cx)];
    float p  = pv ? g[(size_t)sub * (py * 256 + px)] : 0.f;
    float d = p - cv;
    d2 += d * d;
  }
  K[idx] = expf(-0.5f * d2);
}

// ---------------------------------------------------------------------------
// In-place batchnorm over (B,H,W) per channel. One workgroup per channel.
// ---------------------------------------------------------------------------
__global__ void bn_kernel(float* __restrict__ data, const float* __restrict__ gamma,
                          const float* __restrict__ beta, int B, int C, int HW) {
  __shared__ float s1[TPB];
  __shared__ float s2[TPB];
  int c = blockIdx.x;
  int tid = threadIdx.x;
  int N = B * HW;
  float s = 0.f, q = 0.f;
  for (int idx = tid; idx < N; idx += TPB) {
    int b = idx / HW, p = idx - b * HW;
    float v = data[((size_t)b * C + c) * HW + p];
    s += v;
    q += v * v;
  }
  s1[tid] = s; s2[tid] = q;
  __syncthreads();
  for (int o = TPB / 2; o > 0; o >>= 1) {
    if (tid < o) { s1[tid] += s1[tid + o]; s2[tid] += s2[tid + o]; }
    __syncthreads();
  }
  float mean = s1[0] / (float)N;
  float var  = s2[0] / (float)N - mean * mean;
  float sc = rsqrtf(var + 1e-5f) * gamma[c];
  float sh = beta[c] - mean * sc;
  for (int idx = tid; idx < N; idx += TPB) {
    int b = idx / HW, p = idx - b * HW;
    size_t o2 = ((size_t)b * C + c) * HW + p;
    data[o2] = data[o2] * sc + sh;
  }
}

// ---------------------------------------------------------------------------
// Pixel-adaptive convolution via V_WMMA_F32_16X16X4_F32.
//   out[b,o,y,x] = sum_t K[b,t,y,x] * sum_c W[o,c,t] * xin[b,c, pix(y,x,t)]
// MODE 0: stride-2 down conv (pad 1);  MODE 1: transposed conv (zero-stuffed,
// pad (1,2), stride-1 K).  ACT 0/1/2 = none/lrelu/relu on input load.
// FINAL 1 = bias + tanh epilogue (last layer only).
// LDS B tile is pair-interleaved: element (c,n) at [(c>>1)*32 + n*2 + (c&1)]
// so a {k, k+1} B-fragment is a single aligned ds_load_b64.
// ---------------------------------------------------------------------------
template <int MODE, int ACT, int FINAL>
__global__ void __launch_bounds__(TPB)
pac_conv_wmma(const float* __restrict__ xa, const float* __restrict__ xb,
              int Ca, int Cin, int Cout, int H, int W, int OH, int OW,
              const float* __restrict__ Kk, const float* __restrict__ wgt,
              int ws_o, int ws_c, const float* __restrict__ bias,
              float* __restrict__ out) {
  extern __shared__ float bt[];          // [Cin4/2][16][2] K-weighted patch tile
  const int Cin4 = (Cin + 3) & ~3;
  const int Cb   = Cin - Ca;
  const size_t HW = (size_t)H * W;
  const int tid  = threadIdx.x;
  const int lane = tid & 31;
  const int wave = tid >> 5;
  const int tilesX = (OW + 15) >> 4;

  int tile  = blockIdx.x;
  int tx    = tile % tilesX;
  int rem   = tile / tilesX;
  int oy    = rem % OH;
  int b     = rem / OH;
  int xbase = tx << 4;

  int  co_base = (blockIdx.y * WAVES + wave) << 4;
  bool active  = (co_base < Cout);

  v8f acc0 = {0.f, 0.f, 0.f, 0.f, 0.f, 0.f, 0.f, 0.f};
  v8f acc1 = acc0;

  const int m  = lane & 15;          // A row (output channel) / B column (pixel)
  const int kh = (lane >> 4) << 1;   // lanes 0-15 -> K {c,c+1}; 16-31 -> {c+2,c+3}
  int omin = co_base + m;
  if (omin > Cout - 1) omin = Cout - 1;
  const float* wrow = wgt + (size_t)omin * ws_o;

  // Per-thread fixed pixel column for the fill phase.
  const int fn  = tid & 15;          // pixel within tile (invariant: stride TPB)
  const int fc0 = tid >> 4;          // starting channel, step 16
  const int fox = xbase + fn;
  const bool pxok = (fox < OW);
  const size_t baseA0 = (size_t)b * Ca * HW;
  const size_t baseB0 = (size_t)b * (Cb > 0 ? Cb : 1) * HW;
  const int cmain = Cin & ~3;        // quads with no channel masking needed

  for (int t = 0; t < 9; ++t) {
    const int i = t / 3, j = t - 3 * i;

    // ---- per-tap geometry (hoisted out of the channel loop) ----
    bool valid = false;
    size_t pofs = 0;
    if (pxok) {
      if (MODE == 0) {                       // stride-2 down conv, pad 1
        int yi = oy * 2 + i - 1;
        int xi = fox * 2 + j - 1;
        valid = (yi >= 0 && yi < H && xi >= 0 && xi < W);
        if (valid) pofs = (size_t)yi * W + xi;
      } else {                               // transposed conv (zero-stuffed)
        int ys = oy + i - 1, xs = fox + j - 1;
        bool ev = (ys >= 0 && xs >= 0 && !(ys & 1) && !(xs & 1));
        int yi = ys >> 1, xi = xs >> 1;
        valid = ev && (yi < H) && (xi < W);
        if (valid) pofs = (size_t)yi * W + xi;
      }
    }
    float kv = 0.f;
    if (valid) kv = Kk[(((size_t)b * 9 + t) * OH + oy) * OW + fox];

    // ---- cooperative fill of the K-weighted B tile ----
    {
      int c = fc0;
      for (; c < Cin; c += 16) {
        float v = 0.f;
        if (valid) {
          float xv = (c < Ca) ? xa[baseA0 + (size_t)c * HW + pofs]
                              : xb[baseB0 + (size_t)(c - Ca) * HW + pofs];
          if (ACT == 1)      xv = (xv >= 0.f) ? xv : 0.2f * xv;
          else if (ACT == 2) xv = fmaxf(xv, 0.f);
          v = xv * kv;
        }
        bt[((c >> 1) << 5) + (fn << 1) + (c & 1)] = v;
      }
      for (; c < Cin4; c += 16)               // zero-pad rows (Cin % 4 != 0)
        bt[((c >> 1) << 5) + (fn << 1) + (c & 1)] = 0.f;
    }
    __syncthreads();

    // ---- wave-uniform WMMA k-loop over channels for this tap ----
    if (active) {
      const float* wt = wrow + t;
      if (t < 8) __builtin_prefetch(wt + 1, 0, 3);   // next tap's weight row
      int c = 0;
#pragma unroll 2
      for (; c + 8 <= cmain; c += 8) {
        int k0 = c + kh, k1 = c + 4 + kh;
        v2f a0, a1;
        a0.x = wt[(size_t)k0 * ws_c];
        a0.y = wt[(size_t)(k0 + 1) * ws_c];
        a1.x = wt[(size_t)k1 * ws_c];
        a1.y = wt[(size_t)(k1 + 1) * ws_c];
        v2f b0 = *(const v2f*)&bt[((k0 >> 1) << 5) + (m << 1)];
        v2f b1 = *(const v2f*)&bt[((k1 >> 1) << 5) + (m << 1)];
        acc0 = __builtin_amdgcn_wmma_f32_16x16x4_f32(false, a0, false, b0,
                                                     (short)0, acc0, false, false);
        acc1 = __builtin_amdgcn_wmma_f32_16x16x4_f32(false, a1, false, b1,
                                                     (short)0, acc1, false, false);
      }
      for (; c < cmain; c += 4) {
        int k0 = c + kh;
        v2f a0;
        a0.x = wt[(size_t)k0 * ws_c];
        a0.y = wt[(size_t)(k0 + 1) * ws_c];
        v2f b0 = *(const v2f*)&bt[((k0 >> 1) << 5) + (m << 1)];
        acc0 = __builtin_amdgcn_wmma_f32_16x16x4_f32(false, a0, false, b0,
                                                     (short)0, acc0, false, false);
      }
      if (cmain < Cin4) {                      // masked tail (layer 1, Cin=3)
        int k0 = cmain + kh;
        v2f a0;
        a0.x = (k0 < Cin)     ? wt[(size_t)k0 * ws_c]       : 0.f;
        a0.y = (k0 + 1 < Cin) ? wt[(size_t)(k0 + 1) * ws_c] : 0.f;
        v2f b0 = *(const v2f*)&bt[((k0 >> 1) << 5) + (m << 1)];
        acc0 = __builtin_amdgcn_wmma_f32_16x16x4_f32(false, a0, false, b0,
                                                     (short)0, acc0, false, false);
      }
    }
    __syncthreads();
  }

  // ---- epilogue: bias / tanh / masked store (D layout: VGPR v -> M=v, v+8) ----
  if (active) {
    v8f acc = acc0 + acc1;
    int half = lane >> 4;
    int ox = xbase + m;
    if (ox < OW) {
#pragma unroll
      for (int v = 0; v < 8; ++v) {
        int co = co_base + v + 8 * half;
        if (co < Cout) {
          float r = acc[v];
          if (FINAL) {
            r += bias[co];
            r = tanhf(r);
          }
          out[(((size_t)b * Cout + co) * OH + oy) * OW + ox] = r;
        }
      }
    }
  }
}

// ---------------------------------------------------------------------------
// Host orchestration
// ---------------------------------------------------------------------------
extern "C" void kernel_launch(void* const* d_in, const int* in_sizes, int n_in,
                              void* d_out, int out_size, void* d_ws, size_t ws_size,
                              hipStream_t stream) {
  (void)in_sizes; (void)out_size; (void)ws_size;
  const float* in[64];
  for (int i = 0; i < n_in && i < 64; ++i) in[i] = (const float*)d_in[i];

  const float* x    = in[0];
  const float* g_w1 = in[1]; const float* g_b1 = in[2];
  const float* g_w2 = in[3]; const float* g_b2 = in[4];
  const float* g_w3 = in[5]; const float* g_b3 = in[6];
  const float* dw[8]; for (int l = 1; l <= 7; ++l) dw[l] = in[6 + l];      // 7..13
  const float* uw[8]; for (int l = 7; l >= 1; --l) uw[l] = in[21 - l];     // 14..20
  const float* ub1 = in[21];
  const float *dng[8], *dnb[8];
  for (int l = 2; l <= 6; ++l) { dng[l] = in[22 + 2 * (l - 2)]; dnb[l] = in[23 + 2 * (l - 2)]; }
  const float *ung[8], *unb[8];
  for (int l = 2; l <= 7; ++l) { ung[l] = in[32 + 2 * (l - 2)]; unb[l] = in[33 + 2 * (l - 2)]; }

  // ---- carve workspace (floats) ----
  float* ws = (float*)d_ws;
  size_t off = 0;
  auto A = [&](size_t n) { float* p = ws + off; off += n; return p; };
  float* wf    = A((size_t)2 * 12 * 65536);
  float* gt1   = A((size_t)2 * 3 * 65536);
  float* gt2   = A((size_t)2 * 3 * 65536);
  float* guide = A((size_t)2 * 3 * 65536);
  float *Kd[8], *Ku[8];
  for (int l = 1; l <= 7; ++l) { int oh = 128 >> (l - 1); Kd[l] = A((size_t)2 * 9 * oh * oh); }
  for (int l = 1; l <= 7; ++l) { int oh = 256 >> (l - 1); Ku[l] = A((size_t)2 * 9 * oh * oh); }
  const int dC[8] = {0, 64, 128, 256, 512, 512, 512, 512};
  const int uC[8] = {0, 3, 64, 128, 256, 512, 512, 512};
  float* xs_[8];
  for (int l = 1; l <= 7; ++l) { int oh = 128 >> (l - 1); xs_[l] = A((size_t)2 * dC[l] * oh * oh); }
  float* us_[8];
  for (int l = 7; l >= 2; --l) { int oh = 256 >> (l - 1); us_[l] = A((size_t)2 * uC[l] * oh * oh); }

  // ---- guide branch ----
  {
    int n = 2 * 12 * 65536;
    wavelet_kernel<<<(n + TPB - 1) / TPB, TPB, 0, stream>>>(x, wf);
    n = 2 * 3 * 65536;
    conv5_kernel<<<(n + TPB - 1) / TPB, TPB, 0, stream>>>(wf, g_w1, g_b1, gt1, 12, 3, 1);
    conv5_kernel<<<(n + TPB - 1) / TPB, TPB, 0, stream>>>(gt1, g_w2, g_b2, gt2, 3, 3, 1);
    conv5_kernel<<<(n + TPB - 1) / TPB, TPB, 0, stream>>>(gt2, g_w3, g_b3, guide, 3, 3, 0);
  }

  // ---- adapting kernels K per level (down: stride 2 / up: stride 1) ----
  for (int l = 1; l <= 7; ++l) {
    int sub = 1 << (l - 1), Hg = 256 >> (l - 1);
    int ohd = 128 >> (l - 1);
    int n = 2 * 9 * ohd * ohd;
    pac_k_kernel<<<(n + TPB - 1) / TPB, TPB, 0, stream>>>(guide, sub, Hg, 2, ohd, ohd, Kd[l]);
    int ohu = 256 >> (l - 1);
    n = 2 * 9 * ohu * ohu;
    pac_k_kernel<<<(n + TPB - 1) / TPB, TPB, 0, stream>>>(guide, sub, Hg, 1, ohu, ohu, Ku[l]);
  }

  auto launch_pac = [&](const float* pa, const float* pb, int Ca, int Cin, int Cout,
                        int H, int OH, const float* K, const float* w, int ws_o, int ws_c,
                        const float* bias, float* o, int mode, int act, int dt) {
    int tilesX = (OH + 15) / 16;
    dim3 grid((unsigned)(2 * OH * tilesX), (unsigned)((Cout + 16 * WAVES - 1) / (16 * WAVES)));
    int Cin4 = (Cin + 3) & ~3;
    size_t shmem = (size_t)Cin4 * 16 * sizeof(float);
    if (mode == 0 && act == 0)
      pac_conv_wmma<0, 0, 0><<<grid, TPB, shmem, stream>>>(
          pa, pb, Ca, Cin, Cout, H, H, OH, OH, K, w, ws_o, ws_c, bias, o);
    else if (mode == 0)
      pac_conv_wmma<0, 1, 0><<<grid, TPB, shmem, stream>>>(
          pa, pb, Ca, Cin, Cout, H, H, OH, OH, K, w, ws_o, ws_c, bias, o);
    else if (dt)
      pac_conv_wmma<1, 2, 1><<<grid, TPB, shmem, stream>>>(
          pa, pb, Ca, Cin, Cout, H, H, OH, OH, K, w, ws_o, ws_c, bias, o);
    else
      pac_conv_wmma<1, 2, 0><<<grid, TPB, shmem, stream>>>(
          pa, pb, Ca, Cin, Cout, H, H, OH, OH, K, w, ws_o, ws_c, bias, o);
  };

  // ---- encoder: stride-2 PAC convs (+ fused lrelu on input, BN for l=2..6) ----
  const float* prev = x;
  int Cprev = 3;
  for (int l = 1; l <= 7; ++l) {
    int H = 256 >> (l - 1), OHl = 128 >> (l - 1);
    int Co = dC[l];
    launch_pac(prev, nullptr, Cprev, Cprev, Co, H, OHl, Kd[l], dw[l],
               Cprev * 9, 9, nullptr, xs_[l], 0, (l == 1 ? 0 : 1), 0);
    if (l >= 2 && l <= 6)
      bn_kernel<<<Co, TPB, 0, stream>>>(xs_[l], dng[l], dnb[l], 2, Co, OHl * OHl);
    prev = xs_[l];
    Cprev = Co;
  }

  // ---- decoder: transposed PAC convs with fused relu + skip-concat ----
  launch_pac(xs_[7], nullptr, 512, 512, 512, 2, 4, Ku[7], uw[7],
             9, 512 * 9, nullptr, us_[7], 1, 2, 0);
  bn_kernel<<<512, TPB, 0, stream>>>(us_[7], ung[7], unb[7], 2, 512, 16);

  for (int l = 6; l >= 2; --l) {
    int H = 128 >> (l - 1), OHl = 256 >> (l - 1);
    int Ca = dC[l], Cb2 = uC[l + 1], Co = uC[l];
    launch_pac(xs_[l], us_[l + 1], Ca, Ca + Cb2, Co, H, OHl, Ku[l], uw[l],
               9, Co * 9, nullptr, us_[l], 1, 2, 0);
    bn_kernel<<<Co, TPB, 0, stream>>>(us_[l], ung[l], unb[l], 2, Co, OHl * OHl);
  }

  // final layer: skip x1 ++ u2 -> 3 channels, bias + tanh, straight to d_out
  launch_pac(xs_[1], us_[2], 64, 128, 3, 128, 256, Ku[1], uw[1],
             9, 3 * 9, ub1, (float*)d_out, 1, 2, 1);
}